// FeatureExtractor_60997125538021
// MI455X (gfx1250) — compile-verified
//
#include <hip/hip_runtime.h>
#include <hip/hip_bf16.h>

typedef float v2f __attribute__((ext_vector_type(2)));
typedef float v8f __attribute__((ext_vector_type(8)));

#define NCH    256            // channels per level
#define AXMAX  80             // max footprint extent per axis (worst case ~67 incl. align pad)
#define GMAX   1600           // max K-groups: ceil(AXMAX/4)*AXMAX = 20*80
#define WMAXF  (4*GMAX)       // padded weight slots

// Build separable axis weights for one axis of one ROI at one level.
// Exactly mirrors reference prep(): valid = v in [-1, L], clamp at 0,
// lo>=L-1 => lo=hi=L-1, frac=0. Invalid samples contribute zero weight.
__device__ static void axis_weights(float v1, float v2, int L,
                                    float* __restrict__ A, int& base, int& span) {
    float r   = fmaxf(v2 - v1, 1.0f);
    float bin = r * (1.0f / 14.0f);
    for (int i = 0; i < AXMAX; ++i) A[i] = 0.0f;

    int mn = 0x7fffffff, mx = -1;
    for (int i = 0; i < 28; ++i) {
        float off = (float)(i >> 1) + ((float)(i & 1) + 0.5f) * 0.5f;
        float v = v1 + off * bin;
        if (!(v >= -1.0f && v <= (float)L)) continue;
        v = fmaxf(v, 0.0f);
        int lo = (int)floorf(v);
        bool over = (lo >= L - 1);
        if (over) lo = L - 1;
        int hi = over ? (L - 1) : (lo + 1);
        mn = min(mn, lo);
        mx = max(mx, hi);
    }
    if (mx < 0) { base = 0; span = 0; return; }
    base = mn;
    span = mx - mn + 1;
    if (span > AXMAX) span = AXMAX;

    const float inv = 1.0f / 28.0f;
    for (int i = 0; i < 28; ++i) {
        float off = (float)(i >> 1) + ((float)(i & 1) + 0.5f) * 0.5f;
        float v = v1 + off * bin;
        if (!(v >= -1.0f && v <= (float)L)) continue;
        v = fmaxf(v, 0.0f);
        int lo = (int)floorf(v);
        bool over = (lo >= L - 1);
        if (over) lo = L - 1;
        int hi = over ? (L - 1) : (lo + 1);
        float frac = over ? 0.0f : (v - (float)lo);
        int il = lo - mn, ih = hi - mn;
        if (il >= 0 && il < AXMAX) A[il] += (1.0f - frac) * inv;
        if (ih >= 0 && ih < AXMAX) A[ih] += frac * inv;
    }
}

// One block per (roi, level). 256 threads = 8 wave32.
// out[r, l*256 + c] = sum_y Ay[y] * sum_x Ax[x] * feat_l[b, c, y, x]
__global__ __launch_bounds__(256)
void roialign_sep_wmma(const float* __restrict__ f0, const float* __restrict__ f1,
                       const float* __restrict__ f2, const float* __restrict__ f3,
                       const float* __restrict__ boxes, const int* __restrict__ bids,
                       float* __restrict__ out) {
    __shared__ float sAx[AXMAX], sAy[AXMAX];
    __shared__ float sW[WMAXF];    // weights, 4 slots per K-group (zero padded)
    __shared__ int   sB[GMAX];     // per-group base offset into feat plane (8B aligned)
    __shared__ int   sMeta[4];     // x0, y0, Fx, Fy

    const int bx  = blockIdx.x;
    const int r   = bx >> 2;
    const int l   = bx & 3;
    const int tid = threadIdx.x;

    const int Hh = 200 >> l;            // 200,100,50,25
    const int Wl = 320 >> l;            // 320,160,80,40 (all even)
    const int HW = Hh * Wl;             // all divisible by 4
    const float* fp = (l == 0) ? f0 : (l == 1) ? f1 : (l == 2) ? f2 : f3;
    const float  s  = 0.25f / (float)(1 << l);   // exact powers of two

    if (tid == 0) {
        float x1 = boxes[r * 4 + 0] * s;
        float x2 = boxes[r * 4 + 2] * s;
        int x0, Fx;
        axis_weights(x1, x2, Wl, sAx, x0, Fx);
        if ((x0 & 1) && Fx > 0) {       // force even x0 for 8B-aligned row bases
            for (int i = min(Fx, AXMAX - 1); i >= 1; --i) sAx[i] = sAx[i - 1];
            sAx[0] = 0.0f;
            x0 -= 1;
            Fx = min(Fx + 1, AXMAX);
        }
        sMeta[0] = x0; sMeta[2] = Fx;
    } else if (tid == 32) {
        float y1 = boxes[r * 4 + 1] * s;
        float y2 = boxes[r * 4 + 3] * s;
        int y0, Fy;
        axis_weights(y1, y2, Hh, sAy, y0, Fy);
        sMeta[1] = y0; sMeta[3] = Fy;
    }
    __syncthreads();

    const int x0 = sMeta[0], y0 = sMeta[1], Fx = sMeta[2], Fy = sMeta[3];
    const int Gx = (Fx + 3) >> 2;       // groups per row
    const int G  = Fy * Gx;             // total K-groups (0 if ROI degenerates)

    for (int g = tid; g < G; g += 256) {
        int y  = g / Gx;
        int gx = g - y * Gx;
        int xb = gx * 4;
        int base = (y0 + y) * Wl + x0 + xb;     // even
        float ayv = sAy[y];
        float w[4];
        #pragma unroll
        for (int j = 0; j < 4; ++j) {
            int x = xb + j;
            w[j] = (x < Fx) ? ayv * sAx[x] : 0.0f;
        }
        // Clamp tail groups so base+3 stays inside the plane; shift weights to
        // match (dropped high slots are provably zero-weight padding; d is even
        // since base and HW-4 are both even, so alignment is preserved).
        int d = base + 4 - HW;
        if (d < 0) d = 0;
        base -= d;
        sB[g] = base;
        #pragma unroll
        for (int j = 0; j < 4; ++j)
            sW[4 * g + j] = (j >= d) ? w[j - d] : 0.0f;
    }
    __syncthreads();

    // --- WMMA contraction: D(16ch x 16) += A(16ch x 4pix) * B(4pix x 16 replicated) ---
    const int lane = tid & 31;
    const int wv   = tid >> 5;          // 8 waves
    const int cb   = wv * 32;           // 32 channels per wave (2 M-tiles of 16)
    const int b    = bids[r];

    const int ko = (lane < 16) ? 0 : 2; // f32 16x16x4: low lanes K=0,1; high lanes K=2,3
    const float* fb  = fp + (size_t)b * (size_t)NCH * (size_t)HW;
    const float* p0k = fb + (size_t)(cb + (lane & 15)) * (size_t)HW + ko;  // tile 0 plane (+K)
    const float* p1k = p0k + (size_t)16 * (size_t)HW;                      // tile 1 plane (+K)

    v8f acc0 = {0.f,0.f,0.f,0.f,0.f,0.f,0.f,0.f};
    v8f acc1 = {0.f,0.f,0.f,0.f,0.f,0.f,0.f,0.f};

    for (int g = 0; g < G; ++g) {
        int base = sB[g];
        v2f bv = *(const v2f*)&sW[4 * g + ko];    // B: weight pair (ds_load_b64)
        v2f a0 = *(const v2f*)(p0k + base);       // A tile 0: 2 consecutive pixels (b64)
        v2f a1 = *(const v2f*)(p1k + base);       // A tile 1

        acc0 = __builtin_amdgcn_wmma_f32_16x16x4_f32(false, a0, false, bv,
                                                     (short)0, acc0, false, false);
        acc1 = __builtin_amdgcn_wmma_f32_16x16x4_f32(false, a1, false, bv,
                                                     (short)0, acc1, false, false);
    }

    // D layout (32-bit 16x16): VGPR v -> M=v (lanes 0-15) / M=v+8 (lanes 16-31).
    // All N columns identical; extract column 0 from lane 0 and lane 16.
    float* orow = out + (size_t)r * 1024 + (size_t)l * 256;
    if (lane == 0) {
        #pragma unroll
        for (int i = 0; i < 8; ++i) {
            orow[cb + i]      = acc0[i];
            orow[cb + 16 + i] = acc1[i];
        }
    } else if (lane == 16) {
        #pragma unroll
        for (int i = 0; i < 8; ++i) {
            orow[cb + 8 + i]  = acc0[i];
            orow[cb + 24 + i] = acc1[i];
        }
    }
}

extern "C" void kernel_launch(void* const* d_in, const int* in_sizes, int n_in,
                              void* d_out, int out_size, void* d_ws, size_t ws_size,
                              hipStream_t stream) {
    const float* f0    = (const float*)d_in[0];
    const float* f1    = (const float*)d_in[1];
    const float* f2    = (const float*)d_in[2];
    const float* f3    = (const float*)d_in[3];
    const float* boxes = (const float*)d_in[4];
    const int*   bids  = (const int*)d_in[5];
    float*       out   = (float*)d_out;

    const int R = in_sizes[4] / 4;   // boxes is (R,4)
    dim3 grid(R * 4);
    dim3 block(256);
    roialign_sep_wmma<<<grid, block, 0, stream>>>(f0, f1, f2, f3, boxes, bids, out);
}